// TabularPadding_44994077393287
// MI455X (gfx1250) — compile-verified
//
#include <hip/hip_runtime.h>

// Jagged (values, offsets) -> dense padded [B, max_len] (right zero-pad).
// Pure data movement: ~67MB read + 128MB write => HBM-bound (~8.4us at 23.3TB/s).
// Round-1 change: grid-stride over contiguous row chunks (2048 blocks * 32 rows)
// instead of one block per row -- 65536 tiny workgroups would make SPI dispatch
// a first-order cost at this roofline. NT streaming loads, 16B-aligned NT b128
// stores, predicated lanes produce the zero padding.

typedef __attribute__((ext_vector_type(4))) float float4v;

__global__ __launch_bounds__(128) void TabularPadding_gfx1250_kernel(
    const float* __restrict__ values,
    const int*   __restrict__ offsets,
    float*       __restrict__ out,
    int max_len,
    int num_rows,
    int rows_per_block)
{
    const int row_begin = blockIdx.x * rows_per_block;
    int row_end = row_begin + rows_per_block;
    if (row_end > num_rows) row_end = num_rows;

    for (int row = row_begin; row < row_end; ++row) {
        const int start = offsets[row];          // wave-uniform -> s_load_b64 pair
        const int len   = offsets[row + 1] - start;

        const float* __restrict__ src = values + start;
        float*       __restrict__ dst = out + (size_t)row * (size_t)max_len;

        // Stream-ahead prefetch on the read stream (global_prefetch_b8).
        __builtin_prefetch(src + (threadIdx.x << 2) + 1024, /*rw=*/0, /*loc=*/0);

        // Each thread owns 4 consecutive columns per iteration. With max_len=512
        // and 128 threads (4 wave32 waves) the inner loop runs exactly once.
        for (int c = (threadIdx.x << 2); c < max_len; c += (blockDim.x << 2)) {
            float4v v = {0.f, 0.f, 0.f, 0.f};

            if (c + 4 <= len) {
                // Full vector in-range: 4 contiguous NT dword loads
                // (source alignment is arbitrary, so not merged to b128).
                v.x = __builtin_nontemporal_load(src + c + 0);
                v.y = __builtin_nontemporal_load(src + c + 1);
                v.z = __builtin_nontemporal_load(src + c + 2);
                v.w = __builtin_nontemporal_load(src + c + 3);
            } else if (c < len) {
                // Boundary thread: predicated lanes, rest stay zero (padding).
                v.x = __builtin_nontemporal_load(src + c);
                if (c + 1 < len) v.y = __builtin_nontemporal_load(src + c + 1);
                if (c + 2 < len) v.z = __builtin_nontemporal_load(src + c + 2);
            }
            // else: pure padding region -> store zeros.

            if (c + 4 <= max_len) {
                // dst row base 16B aligned, c % 4 == 0 -> NT global_store_b128.
                __builtin_nontemporal_store(v, (float4v*)(dst + c));
            } else {
                // Generic tail guard (not taken for max_len=512).
                if (c + 0 < max_len) __builtin_nontemporal_store(v.x, dst + c + 0);
                if (c + 1 < max_len) __builtin_nontemporal_store(v.y, dst + c + 1);
                if (c + 2 < max_len) __builtin_nontemporal_store(v.z, dst + c + 2);
            }
        }
    }
}

extern "C" void kernel_launch(void* const* d_in, const int* in_sizes, int n_in,
                              void* d_out, int out_size, void* d_ws, size_t ws_size,
                              hipStream_t stream) {
    (void)n_in; (void)d_ws; (void)ws_size;

    const float* values  = (const float*)d_in[0];
    const int*   offsets = (const int*)d_in[1];   // offsets fit int32 (~16.8M tokens)

    const int num_rows = in_sizes[1] - 1;         // 65536
    const int max_len  = out_size / num_rows;     // 512

    float* out = (float*)d_out;

    // Amortize workgroup launch: each block owns a contiguous chunk of rows.
    const int rows_per_block = 32;                // 2048 blocks, 8192 waves total
    const int num_blocks = (num_rows + rows_per_block - 1) / rows_per_block;

    dim3 grid((unsigned)num_blocks);
    dim3 block(128);   // 4 wave32 waves; 128 threads x 4 cols = 512 columns/iter
    TabularPadding_gfx1250_kernel<<<grid, block, 0, stream>>>(
        values, offsets, out, max_len, num_rows, rows_per_block);
}